// CoGNN_18184891531339
// MI455X (gfx1250) — compile-verified
//
#include <hip/hip_runtime.h>

// ---------------------------------------------------------------------------
// MTGNN-style forward pass for gfx1250 (MI455X).
// All dense GEMMs (adjacency build, graph propagation, channel mix) run on
// V_WMMA_F32_16X16X4_F32. One wave owns 16x16 (adj/mix) or 16x64 (spatial)
// output tiles. Inner loops are branch-free (zero-padded operands, b64 loads
// for K-contiguous data); epilogues take a uniform fast path on full tiles.
// Spatial K loop kept at unroll 2 so the hot loop stays I-cache resident.
// ---------------------------------------------------------------------------

#define BN   32      // batch
#define CC   32      // channels
#define NV   1000    // graph nodes
#define NVP  1008    // padded node rows for A matrices (zero rows 1000..1007)
#define KTOP 20
#define ALPHA_F 0.05f
#define EPS_F 1e-5f

typedef __attribute__((ext_vector_type(2))) float v2f;
typedef __attribute__((ext_vector_type(8))) float v8f;

// ------------------------- input prep --------------------------------------

__global__ void k_inp(const float* __restrict__ xin, float* __restrict__ inp)
{
    int idx = blockIdx.x * blockDim.x + threadIdx.x;
    if (idx >= BN * NV * 19) return;
    int l = idx % 19;
    int v = (idx / 19) % NV;
    int n = idx / (19 * NV);
    float val = 0.f;
    if (l >= 7) {
        int t  = l - 7;
        int vv = (v < 500) ? v : v - 500;
        int ch = (v < 500) ? 0 : 1;
        val = xin[(((size_t)n * 12 + t) * 500 + vv) * 4 + ch];
    }
    inp[idx] = val;
}

__global__ void k_start(const float* __restrict__ inp, const float* __restrict__ sw,
                        const float* __restrict__ sb, float* __restrict__ x)
{
    size_t idx = (size_t)blockIdx.x * blockDim.x + threadIdx.x;
    if (idx >= (size_t)BN * CC * NV * 19) return;
    int l = idx % 19;
    int v = (idx / 19) % NV;
    int c = (idx / ((size_t)19 * NV)) % CC;
    int n = idx / ((size_t)19 * NV * CC);
    x[idx] = inp[((size_t)n * NV + v) * 19 + l] * sw[c] + sb[c];
}

__global__ void k_skip0(const float* __restrict__ inp, const float* __restrict__ s0w,
                        const float* __restrict__ s0b, float* __restrict__ skip)
{
    int idx = blockIdx.x * blockDim.x + threadIdx.x;
    if (idx >= BN * 64 * NV) return;
    int v = idx % NV;
    int k = (idx / NV) % 64;
    int n = idx / (NV * 64);
    const float* ip = inp + ((size_t)n * NV + v) * 19;
    float acc = s0b[k];
    for (int l = 0; l < 19; ++l) acc += ip[l] * s0w[k * 19 + l];
    skip[idx] = acc;
}

// ------------------------- adjacency ---------------------------------------

// nv buffers padded to 512 rows per mixture; rows >= 500 are zero.
__global__ void k_nv(const float* __restrict__ e, const float* __restrict__ lw,
                     const float* __restrict__ lb, float* __restrict__ nv)
{
    int idx = blockIdx.x * blockDim.x + threadIdx.x;
    if (idx >= 4 * 512 * 40) return;
    int o  = idx % 40;
    int nn = (idx / 40) % 512;
    int m  = idx / (40 * 512);
    float acc = 0.f;
    if (nn < 500) {
        const float* ev = e  + ((size_t)m * 500 + nn) * 40;
        const float* wv = lw + ((size_t)m * 40 + o) * 40;
        acc = lb[m * 40 + o];
        for (int d = 0; d < 40; ++d) acc += ev[d] * wv[d];
    }
    nv[idx] = acc;      // layout [m][512][40]
}

// a[m] = nv1[m] (500x40) x nv2[m]^T (40x500), written into 1000x1000 adj.
// Loads are unguarded (rows padded to 512); stores take a uniform fast path
// on interior tiles.
__global__ __launch_bounds__(32)
void k_adj_gemm(const float* __restrict__ nv1, const float* __restrict__ nv2,
                float* __restrict__ adj)
{
    const int m    = blockIdx.z;
    const int ri   = m >> 1, rj = m & 1;
    const int lane = threadIdx.x;
    const int half = lane >> 4, r = lane & 15;
    const int n0 = blockIdx.x * 16;
    const int k0 = blockIdx.y * 16;
    const int arow = n0 + r;                 // < 512 (padded)
    const int bcol = k0 + r;                 // < 512 (padded)
    const float2* Ap = (const float2*)(nv1 + (size_t)m * 512 * 40 + (size_t)arow * 40) + half;
    const float2* Bp = (const float2*)(nv2 + (size_t)m * 512 * 40 + (size_t)bcol * 40) + half;
    v8f acc = {0.f,0.f,0.f,0.f,0.f,0.f,0.f,0.f};
#pragma unroll
    for (int d0 = 0; d0 < 40; d0 += 4) {
        float2 av = Ap[d0 >> 1];             // nv1[arow][d0+2*half .. +1]
        float2 bv = Bp[d0 >> 1];
        v2f a, b;
        a[0] = av.x; a[1] = av.y;
        b[0] = bv.x; b[1] = bv.y;
        acc = __builtin_amdgcn_wmma_f32_16x16x4_f32(false, a, false, b,
                                                    (short)0, acc, false, false);
    }
    float* op = adj + (size_t)(ri * 500 + n0 + half * 8) * NV + rj * 500 + k0 + r;
    if (n0 + 16 <= 500 && k0 + 16 <= 500) {     // uniform fast path
#pragma unroll
        for (int i = 0; i < 8; ++i) { op[0] = acc[i]; op += NV; }
    } else {
        const bool cok = (k0 + r) < 500;
#pragma unroll
        for (int i = 0; i < 8; ++i) {
            if (cok && (n0 + i + half * 8) < 500) op[0] = acc[i];
            op += NV;
        }
    }
}

__global__ __launch_bounds__(128)
void k_topk(float* __restrict__ adj)
{
    __shared__ float sv[NV];
    __shared__ int   sel[KTOP];
    __shared__ float rmax[128];
    __shared__ int   rmaxi[128];
    const int v = blockIdx.x;
    float* row = adj + (size_t)v * NV;
    for (int j = threadIdx.x; j < NV; j += 128) sv[j] = row[j];
    __syncthreads();
    for (int t = 0; t < KTOP; ++t) {
        float bm = -3.4e38f; int bi = 0;
        for (int j = threadIdx.x; j < NV; j += 128) {
            float x = sv[j];
            if (x > bm) { bm = x; bi = j; }
        }
        rmax[threadIdx.x] = bm; rmaxi[threadIdx.x] = bi;
        __syncthreads();
        for (int s = 64; s > 0; s >>= 1) {
            if (threadIdx.x < s && rmax[threadIdx.x + s] > rmax[threadIdx.x]) {
                rmax[threadIdx.x]  = rmax[threadIdx.x + s];
                rmaxi[threadIdx.x] = rmaxi[threadIdx.x + s];
            }
            __syncthreads();
        }
        if (threadIdx.x == 0) { sel[t] = rmaxi[0]; sv[rmaxi[0]] = -3.4e38f; }
        __syncthreads();
    }
    for (int j = threadIdx.x; j < NV; j += 128) {
        bool keep = false;
        for (int t = 0; t < KTOP; ++t) keep |= (sel[t] == j);
        if (!keep) row[j] = 0.f;
    }
}

__global__ void k_deg(const float* __restrict__ adj, float* __restrict__ invd)
{
    int v = blockIdx.x * blockDim.x + threadIdx.x;
    if (v >= NV) return;
    float rs = 0.f, cs = 0.f;
    for (int w = 0; w < NV; ++w) {
        rs += adj[(size_t)v * NV + w];
        cs += adj[(size_t)w * NV + v];
    }
    invd[v]      = 1.f / (1.f + rs);
    invd[NV + v] = 1.f / (1.f + cs);
}

// A1/A2 are NVP x NV; rows >= NV are zero so GEMM row loads need no guard.
__global__ void k_buildA(const float* __restrict__ adj, const float* __restrict__ invd,
                         float* __restrict__ A1, float* __restrict__ A2)
{
    int idx = blockIdx.x * blockDim.x + threadIdx.x;
    if (idx >= NVP * NV) return;
    int w = idx % NV, v = idx / NV;
    if (v < NV) {
        float diag = (v == w) ? 1.f : 0.f;
        A1[idx] = (adj[(size_t)v * NV + w] + diag) * invd[v];
        A2[idx] = (adj[(size_t)w * NV + v] + diag) * invd[NV + v];
    } else {
        A1[idx] = 0.f;
        A2[idx] = 0.f;
    }
}

// ------------------------- mixprop -----------------------------------------

__global__ void k_initacc(float* __restrict__ acc, const float* __restrict__ b1,
                          const float* __restrict__ b2, int L)
{
    size_t idx = (size_t)blockIdx.x * blockDim.x + threadIdx.x;
    if (idx >= (size_t)BN * CC * NV * L) return;
    int o = (idx / ((size_t)NV * L)) % CC;
    acc[idx] = b1[o] + b2[o];
}

// S[v, j] = sum_w A[v,w] * H[w, j],  H layout [nc][w][l], j = nc*L + l.
// One wave: 16 rows x 64 columns (4 tiles); A pair loaded once per K step and
// reused for 4 WMMAs. All loads unguarded (A rows padded, J % 64 == 0).
__global__ __launch_bounds__(32)
void k_spatial_gemm(const float* __restrict__ A, const float* __restrict__ H,
                    float* __restrict__ S, int L)
{
    const int lane = threadIdx.x;
    const int half = lane >> 4, r = lane & 15;
    const int v0 = blockIdx.x * 16;
    const int arow = v0 + r;                         // < NVP, padded
    const float2* Ap = (const float2*)(A + (size_t)arow * NV) + half;
    const int j0 = blockIdx.y * 64 + r;
    size_t colOff[4];
    v8f acc[4];
#pragma unroll
    for (int t = 0; t < 4; ++t) {
        int j = j0 + t * 16;
        int c = j / L, l = j - c * L;
        colOff[t] = (size_t)c * NV * L + l;
        acc[t] = (v8f){0.f,0.f,0.f,0.f,0.f,0.f,0.f,0.f};
    }
    const float* H0 = H + (size_t)(half * 2) * L;    // k = w0 + 2*half
#pragma unroll 2
    for (int w0 = 0; w0 < NV; w0 += 4) {
        float2 av = Ap[w0 >> 1];                     // A[arow][w0+2h], [w0+2h+1]
        v2f a; a[0] = av.x; a[1] = av.y;
#pragma unroll
        for (int t = 0; t < 4; ++t) {
            const float* hp = H0 + colOff[t] + (size_t)w0 * L;
            v2f b; b[0] = hp[0]; b[1] = hp[L];
            acc[t] = __builtin_amdgcn_wmma_f32_16x16x4_f32(false, a, false, b,
                                                           (short)0, acc[t], false, false);
        }
    }
    if (v0 + 16 <= NV) {                             // uniform fast path
#pragma unroll
        for (int t = 0; t < 4; ++t) {
            float* sp = S + colOff[t] + (size_t)(v0 + half * 8) * L;
#pragma unroll
            for (int i = 0; i < 8; ++i) { sp[0] = acc[t][i]; sp += L; }
        }
    } else {                                         // edge row tile
#pragma unroll
        for (int t = 0; t < 4; ++t) {
            float* sp = S + colOff[t] + (size_t)(v0 + half * 8) * L;
#pragma unroll
            for (int i = 0; i < 8; ++i) {
                if (v0 + half * 8 + i < NV) sp[0] = acc[t][i];
                sp += L;
            }
        }
    }
}

__global__ void k_combine(const float* __restrict__ x, const float* __restrict__ s,
                          float* __restrict__ h, size_t nelem)
{
    size_t idx = (size_t)blockIdx.x * blockDim.x + threadIdx.x;
    if (idx >= nelem) return;
    h[idx] = ALPHA_F * x[idx] + (1.f - ALPHA_F) * s[idx];
}

// acc[n][o][j] += W[o, kblk*32 + c] * H[n][c][j]  (o=32, c=32, j=V*L).
// Loads unguarded (big buffers over-allocated by 32 floats); stores take a
// uniform fast path except on the final column tile.
__global__ __launch_bounds__(32)
void k_mix_gemm(const float* __restrict__ Wf, int kblk,
                const float* __restrict__ H, float* __restrict__ acc, int L)
{
    const int lane = threadIdx.x;
    const int half = lane >> 4, r = lane & 15;
    const int n  = blockIdx.z;
    const int m0 = blockIdx.y * 16;
    const int j0 = blockIdx.x * 16;
    const int VL = NV * L;
    const int myj = j0 + r;
    const float* Hn = H   + (size_t)n * CC * VL + myj;
    float*       an = acc + (size_t)n * CC * VL + myj;
    v8f cacc;
#pragma unroll
    for (int i = 0; i < 8; ++i)
        cacc[i] = an[(size_t)(m0 + i + half * 8) * VL];
    const float2* Wp = (const float2*)(Wf + (size_t)(m0 + r) * 96 + kblk * 32) + half;
#pragma unroll
    for (int k0 = 0; k0 < 32; k0 += 4) {
        const int k = k0 + half * 2;
        float2 wv = Wp[k0 >> 1];
        v2f a, b;
        a[0] = wv.x; a[1] = wv.y;
        b[0] = Hn[(size_t)k * VL];
        b[1] = Hn[(size_t)(k + 1) * VL];
        cacc = __builtin_amdgcn_wmma_f32_16x16x4_f32(false, a, false, b,
                                                     (short)0, cacc, false, false);
    }
    float* op = an + (size_t)(m0 + half * 8) * VL;
    if (j0 + 16 <= VL) {                             // uniform fast path
#pragma unroll
        for (int i = 0; i < 8; ++i) { op[0] = cacc[i]; op += VL; }
    } else if (myj < VL) {                           // tail column tile
#pragma unroll
        for (int i = 0; i < 8; ++i) { op[0] = cacc[i]; op += VL; }
    }
}

// ------------------------- temporal block ----------------------------------

__global__ void k_inception(const float* __restrict__ xin, float* __restrict__ xout,
                            const float* __restrict__ fw2, const float* __restrict__ fw3,
                            const float* __restrict__ fw6, const float* __restrict__ fw7,
                            const float* __restrict__ fb,
                            const float* __restrict__ gw2, const float* __restrict__ gw3,
                            const float* __restrict__ gw6, const float* __restrict__ gw7,
                            const float* __restrict__ gb,
                            int Lin, int Lout)
{
    size_t idx = (size_t)blockIdx.x * blockDim.x + threadIdx.x;
    if (idx >= (size_t)BN * CC * NV * Lout) return;
    int t  = idx % Lout;
    int v  = (idx / Lout) % NV;
    int co = (idx / ((size_t)Lout * NV)) % CC;
    int n  = idx / ((size_t)Lout * NV * CC);
    int g   = co >> 3;
    int co8 = co & 7;
    int k = (g == 0) ? 2 : (g == 1) ? 3 : (g == 2) ? 6 : 7;
    const float* fw = (g == 0) ? fw2 : (g == 1) ? fw3 : (g == 2) ? fw6 : fw7;
    const float* gw = (g == 0) ? gw2 : (g == 1) ? gw3 : (g == 2) ? gw6 : gw7;
    float f  = fb[co];
    float ga = gb[co];
    const int tbase = t + 7 - k;
    for (int c = 0; c < CC; ++c) {
        const float* xc  = xin + (((size_t)n * CC + c) * NV + v) * Lin + tbase;
        const float* fwc = fw + ((size_t)co8 * CC + c) * k;
        const float* gwc = gw + ((size_t)co8 * CC + c) * k;
        for (int dt = 0; dt < k; ++dt) {
            float xv = xc[dt];
            f  += xv * fwc[dt];
            ga += xv * gwc[dt];
        }
    }
    float s = 1.f / (1.f + __expf(-ga));
    xout[idx] = tanhf(f) * s;
}

__global__ void k_skipconv(const float* __restrict__ x, const float* __restrict__ sw,
                           const float* __restrict__ sb, float* __restrict__ skip, int L)
{
    int idx = blockIdx.x * blockDim.x + threadIdx.x;
    if (idx >= BN * 64 * NV) return;
    int v = idx % NV;
    int k = (idx / NV) % 64;
    int n = idx / (NV * 64);
    float acc = sb[k];
    for (int c = 0; c < CC; ++c) {
        const float* xc = x  + (((size_t)n * CC + c) * NV + v) * L;
        const float* wc = sw + ((size_t)k * CC + c) * L;
        for (int l = 0; l < L; ++l) acc += xc[l] * wc[l];
    }
    skip[idx] += acc;
}

__global__ void k_resadd(float* __restrict__ x, const float* __restrict__ res,
                         int Lin, int Lout)
{
    size_t idx = (size_t)blockIdx.x * blockDim.x + threadIdx.x;
    if (idx >= (size_t)BN * CC * NV * Lout) return;
    int t = idx % Lout;
    size_t p = idx / Lout;           // (n*C + c)*V + v
    x[idx] += res[p * Lin + (Lin - Lout) + t];
}

__global__ __launch_bounds__(256)
void k_lnreduce(const float* __restrict__ x, float* __restrict__ stats, int CVL)
{
    __shared__ float ssum[256], ssq[256];
    const int n = blockIdx.x;
    const float* xn = x + (size_t)n * CVL;
    float s = 0.f, q = 0.f;
    for (int i = threadIdx.x; i < CVL; i += 256) {
        float v = xn[i]; s += v; q += v * v;
    }
    ssum[threadIdx.x] = s; ssq[threadIdx.x] = q;
    __syncthreads();
    for (int st = 128; st > 0; st >>= 1) {
        if (threadIdx.x < st) {
            ssum[threadIdx.x] += ssum[threadIdx.x + st];
            ssq[threadIdx.x]  += ssq[threadIdx.x + st];
        }
        __syncthreads();
    }
    if (threadIdx.x == 0) {
        float mu  = ssum[0] / CVL;
        float var = ssq[0] / CVL - mu * mu;
        stats[2 * n]     = mu;
        stats[2 * n + 1] = rsqrtf(var + EPS_F);
    }
}

__global__ void k_lnapply(float* __restrict__ x, const float* __restrict__ stats,
                          const float* __restrict__ nw, const float* __restrict__ nb, int L)
{
    size_t idx = (size_t)blockIdx.x * blockDim.x + threadIdx.x;
    if (idx >= (size_t)BN * CC * NV * L) return;
    int t = idx % L;
    int v = (idx / L) % NV;
    int c = (idx / ((size_t)L * NV)) % CC;
    int n = idx / ((size_t)L * NV * CC);
    float mu = stats[2 * n], inv = stats[2 * n + 1];
    size_t wi = ((size_t)c * NV + v) * L + t;
    x[idx] = (x[idx] - mu) * inv * nw[wi] + nb[wi];
}

// ------------------------- head --------------------------------------------

__global__ __launch_bounds__(128)
void k_head(const float* __restrict__ x, const float* __restrict__ skip,
            const float* __restrict__ sew, const float* __restrict__ seb,
            const float* __restrict__ e1w, const float* __restrict__ e1b,
            const float* __restrict__ e2w, const float* __restrict__ e2b,
            float* __restrict__ out)
{
    __shared__ float s64[64];
    __shared__ float y[128];
    const int v = blockIdx.x % NV;
    const int n = blockIdx.x / NV;
    const int tid = threadIdx.x;
    if (tid < 64) {
        float acc = skip[((size_t)n * 64 + tid) * NV + v] + seb[tid];
        for (int c = 0; c < CC; ++c)
            acc += x[((size_t)n * CC + c) * NV + v] * sew[tid * CC + c];
        s64[tid] = fmaxf(acc, 0.f);
    }
    __syncthreads();
    {
        float acc = e1b[tid];
        for (int k = 0; k < 64; ++k) acc += s64[k] * e1w[tid * 64 + k];
        y[tid] = fmaxf(acc, 0.f);
    }
    __syncthreads();
    if (tid < 12) {
        float acc = e2b[tid];
        for (int j = 0; j < 128; ++j) acc += y[j] * e2w[tid * 128 + j];
        out[((size_t)n * 12 + tid) * NV + v] = acc;
    }
}

// ------------------------- host orchestration -------------------------------

extern "C" void kernel_launch(void* const* d_in, const int* in_sizes, int n_in,
                              void* d_out, int out_size, void* d_ws, size_t ws_size,
                              hipStream_t stream)
{
    (void)in_sizes; (void)n_in; (void)out_size; (void)ws_size;

    const float* x_in    = (const float*)d_in[0];
    const float* start_w = (const float*)d_in[1];
    const float* start_b = (const float*)d_in[2];
    const float* skip0_w = (const float*)d_in[3];
    const float* skip0_b = (const float*)d_in[4];
    const float* emb1    = (const float*)d_in[5];
    const float* emb2    = (const float*)d_in[6];
    const float* lin1_w  = (const float*)d_in[7];
    const float* lin1_b  = (const float*)d_in[8];
    const float* lin2_w  = (const float*)d_in[9];
    const float* lin2_b  = (const float*)d_in[10];
    const float* g1_w    = (const float*)d_in[11];
    const float* g1_b    = (const float*)d_in[12];
    const float* g2_w    = (const float*)d_in[13];
    const float* g2_b    = (const float*)d_in[14];
    const float* filt_w[4] = {(const float*)d_in[15], (const float*)d_in[16],
                              (const float*)d_in[17], (const float*)d_in[18]};
    const float* filt_b  = (const float*)d_in[19];
    const float* gate_w[4] = {(const float*)d_in[20], (const float*)d_in[21],
                              (const float*)d_in[22], (const float*)d_in[23]};
    const float* gate_b  = (const float*)d_in[24];
    const float* skip_w[3] = {(const float*)d_in[25], (const float*)d_in[29], (const float*)d_in[33]};
    const float* skip_b[3] = {(const float*)d_in[26], (const float*)d_in[30], (const float*)d_in[34]};
    const float* norm_w[3] = {(const float*)d_in[27], (const float*)d_in[31], (const float*)d_in[35]};
    const float* norm_b[3] = {(const float*)d_in[28], (const float*)d_in[32], (const float*)d_in[36]};
    const float* skipE_w = (const float*)d_in[37];
    const float* skipE_b = (const float*)d_in[38];
    const float* end1_w  = (const float*)d_in[39];
    const float* end1_b  = (const float*)d_in[40];
    const float* end2_w  = (const float*)d_in[41];
    const float* end2_b  = (const float*)d_in[42];

    // workspace layout (floats); needs ~335 MB
    const size_t BIG = (size_t)BN * CC * NV * 19 + 32;   // +32 pad for unguarded tails
    float* ws = (float*)d_ws;
    size_t off = 0;
    float* inp   = ws + off; off += (size_t)BN * NV * 19;
    float* bufA  = ws + off; off += BIG;   // x / hop ping
    float* bufB  = ws + off; off += BIG;   // hop pong / layer output
    float* sbuf  = ws + off; off += BIG;   // spatial GEMM output / h1
    float* accb  = ws + off; off += BIG;   // mixprop accumulator
    float* skipb = ws + off; off += (size_t)BN * 64 * NV;
    float* adj   = ws + off; off += (size_t)NV * NV;
    float* A1    = ws + off; off += (size_t)NVP * NV;
    float* A2    = ws + off; off += (size_t)NVP * NV;
    float* nv1   = ws + off; off += 4ull * 512 * 40;
    float* nv2   = ws + off; off += 4ull * 512 * 40;
    float* invd  = ws + off; off += 2 * NV;
    float* stats = ws + off; off += 64;

    auto grid = [](size_t n) { return (unsigned)((n + 255) / 256); };

    float* xcur  = bufA;
    float* hfree = bufB;

    // prologue
    k_inp  <<<grid((size_t)BN * NV * 19), 256, 0, stream>>>(x_in, inp);
    k_start<<<grid((size_t)BN * CC * NV * 19), 256, 0, stream>>>(inp, start_w, start_b, xcur);
    k_skip0<<<grid((size_t)BN * 64 * NV), 256, 0, stream>>>(inp, skip0_w, skip0_b, skipb);

    const int Lin_a[3]  = {19, 13, 7};
    const int Lout_a[3] = {13, 7, 1};
    const int kflt[4]   = {2, 3, 6, 7};

    for (int i = 0; i < 3; ++i) {
        const int L  = Lin_a[i];
        const int Lo = Lout_a[i];
        const size_t nel  = (size_t)BN * CC * NV * L;
        const size_t nelo = (size_t)BN * CC * NV * Lo;

        // --- adjacency ---
        k_nv<<<grid(4 * 512 * 40), 256, 0, stream>>>(emb1 + i * 80000, lin1_w + i * 6400,
                                                     lin1_b + i * 160, nv1);
        k_nv<<<grid(4 * 512 * 40), 256, 0, stream>>>(emb2 + i * 80000, lin2_w + i * 6400,
                                                     lin2_b + i * 160, nv2);
        k_adj_gemm<<<dim3(32, 32, 4), 32, 0, stream>>>(nv1, nv2, adj);
        k_topk<<<NV, 128, 0, stream>>>(adj);
        k_deg<<<grid(NV), 256, 0, stream>>>(adj, invd);
        k_buildA<<<grid((size_t)NVP * NV), 256, 0, stream>>>(adj, invd, A1, A2);

        // --- mixprop (both directions), accumulate channel mix into accb ---
        k_initacc<<<grid(nel), 256, 0, stream>>>(accb, g1_b + i * 32, g2_b + i * 32, L);
        const unsigned vt = NVP / 16;                      // 63 row tiles
        const unsigned jt = (unsigned)((1024 * L) / 64);   // 4-wide column tiles
        const unsigned mt = (unsigned)((NV * L + 15) / 16);
        for (int br = 0; br < 2; ++br) {
            const float* Aadj = br ? A2 : A1;
            const float* Wm   = (br ? g2_w : g1_w) + i * 32 * 96;
            // hop 0 contribution (h0 = x)
            k_mix_gemm<<<dim3(mt, 2, BN), 32, 0, stream>>>(Wm, 0, xcur, accb, L);
            // hop 1
            k_spatial_gemm<<<dim3(vt, jt), 32, 0, stream>>>(Aadj, xcur, sbuf, L);
            k_combine<<<grid(nel), 256, 0, stream>>>(xcur, sbuf, sbuf, nel);
            k_mix_gemm<<<dim3(mt, 2, BN), 32, 0, stream>>>(Wm, 1, sbuf, accb, L);
            // hop 2
            k_spatial_gemm<<<dim3(vt, jt), 32, 0, stream>>>(Aadj, sbuf, hfree, L);
            k_combine<<<grid(nel), 256, 0, stream>>>(xcur, hfree, hfree, nel);
            k_mix_gemm<<<dim3(mt, 2, BN), 32, 0, stream>>>(Wm, 2, hfree, accb, L);
        }

        // --- dilated inception + gating -> hfree (new x, length Lo) ---
        k_inception<<<grid(nelo), 256, 0, stream>>>(
            accb, hfree,
            filt_w[0] + i * 8 * 32 * kflt[0], filt_w[1] + i * 8 * 32 * kflt[1],
            filt_w[2] + i * 8 * 32 * kflt[2], filt_w[3] + i * 8 * 32 * kflt[3],
            filt_b + i * 32,
            gate_w[0] + i * 8 * 32 * kflt[0], gate_w[1] + i * 8 * 32 * kflt[1],
            gate_w[2] + i * 8 * 32 * kflt[2], gate_w[3] + i * 8 * 32 * kflt[3],
            gate_b + i * 32, L, Lo);

        // --- skip, residual, layernorm ---
        k_skipconv<<<grid((size_t)BN * 64 * NV), 256, 0, stream>>>(hfree, skip_w[i],
                                                                   skip_b[i], skipb, Lo);
        k_resadd<<<grid(nelo), 256, 0, stream>>>(hfree, xcur, L, Lo);
        k_lnreduce<<<BN, 256, 0, stream>>>(hfree, stats, CC * NV * Lo);
        k_lnapply<<<grid(nelo), 256, 0, stream>>>(hfree, stats, norm_w[i], norm_b[i], Lo);

        // swap: hfree becomes the new x, old x buffer becomes scratch
        float* tswap = xcur; xcur = hfree; hfree = tswap;
    }

    // epilogue: skipE + relu + end1 + relu + end2
    k_head<<<BN * NV, 128, 0, stream>>>(xcur, skipb, skipE_w, skipE_b,
                                        end1_w, end1_b, end2_w, end2_b, (float*)d_out);
}